// DWActionValueModel_3556232922579
// MI455X (gfx1250) — compile-verified
//
#include <hip/hip_runtime.h>
#include <math.h>

#define N_NODES    100000
#define N_EDGES    3200000
#define N_FEAT     8
#define NUM_GRAPHS 64
#define LEAKY      0.2f
#define EPS_NORM   1e-5f

typedef float v2f   __attribute__((ext_vector_type(2)));
typedef float v8f   __attribute__((ext_vector_type(8)));
typedef float fvec4 __attribute__((ext_vector_type(4)));
typedef int   ivec4 __attribute__((ext_vector_type(4)));

#ifndef __has_builtin
#define __has_builtin(x) 0
#endif
#if __has_builtin(__builtin_amdgcn_wmma_f32_16x16x4_f32)
#define HAVE_WMMA_F32X4 1
#else
#define HAVE_WMMA_F32X4 0
#endif

// ---- order-preserving float<->int key (for atomicMax-based segment max) ----
__device__ __forceinline__ int f2key(float f) {
    int i = __float_as_int(f);
    return i ^ ((i >> 31) & 0x7fffffff);
}
__device__ __forceinline__ float key2f(int k) {
    return __int_as_float(k ^ ((k >> 31) & 0x7fffffff));
}

__device__ __forceinline__ float dot8(const fvec4 a0, const fvec4 a1,
                                      const fvec4 b0, const fvec4 b1) {
    float acc;
    acc = a0.x * b0.x;
    acc = fmaf(a0.y, b0.y, acc);
    acc = fmaf(a0.z, b0.z, acc);
    acc = fmaf(a0.w, b0.w, acc);
    acc = fmaf(a1.x, b1.x, acc);
    acc = fmaf(a1.y, b1.y, acc);
    acc = fmaf(a1.z, b1.z, acc);
    acc = fmaf(a1.w, b1.w, acc);
    return acc;
}

// ---- small init: zero stats / graph-pool accumulators ----
__global__ void k_init_small(float* __restrict__ stats, float* __restrict__ gsum,
                             float* __restrict__ gcnt) {
    int t = threadIdx.x;                 // launched with 512 threads
    if (t < 16) stats[t] = 0.f;
    if (t < NUM_GRAPHS * N_FEAT) gsum[t] = 0.f;
    if (t < NUM_GRAPHS) gcnt[t] = 0.f;
}

// ---- per-feature sum / sumsq over all nodes (LDS ds_add_f32 + few global atomics) ----
__global__ __launch_bounds__(256) void k_stats(const float* __restrict__ xin,
                                               float* __restrict__ stats, int n) {
    __shared__ float ls[16];
    if (threadIdx.x < 16) ls[threadIdx.x] = 0.f;
    __syncthreads();
    int i = blockIdx.x * blockDim.x + threadIdx.x;
    if (i < n) {
        const fvec4 v0 = *(const fvec4*)(xin + i * N_FEAT);
        const fvec4 v1 = *(const fvec4*)(xin + i * N_FEAT + 4);
        atomicAdd(&ls[0], v0.x);  atomicAdd(&ls[8],  v0.x * v0.x);
        atomicAdd(&ls[1], v0.y);  atomicAdd(&ls[9],  v0.y * v0.y);
        atomicAdd(&ls[2], v0.z);  atomicAdd(&ls[10], v0.z * v0.z);
        atomicAdd(&ls[3], v0.w);  atomicAdd(&ls[11], v0.w * v0.w);
        atomicAdd(&ls[4], v1.x);  atomicAdd(&ls[12], v1.x * v1.x);
        atomicAdd(&ls[5], v1.y);  atomicAdd(&ls[13], v1.y * v1.y);
        atomicAdd(&ls[6], v1.z);  atomicAdd(&ls[14], v1.z * v1.z);
        atomicAdd(&ls[7], v1.w);  atomicAdd(&ls[15], v1.w * v1.w);
    }
    __syncthreads();
    if (threadIdx.x < 16) {
        float v = ls[threadIdx.x];
        if (v != 0.f) atomicAdd(&stats[threadIdx.x], v);
    }
}

// ---- x = (in - mu) / (std + eps); 4 elements/thread, b128 path ----
__global__ __launch_bounds__(256) void k_dwnorm(const float* __restrict__ xin,
                                                const float* __restrict__ stats,
                                                float* __restrict__ x, int total4) {
    int t = blockIdx.x * blockDim.x + threadIdx.x;
    if (t >= total4) return;
    const int base = t * 4;
    const int f0 = base & (N_FEAT - 1);         // 0 or 4
    const float inv_n = 1.0f / (float)N_NODES;
    fvec4 v = *(const fvec4*)(xin + base);
    fvec4 r;
    {
        float mu = stats[f0 + 0] * inv_n;
        float sd = sqrtf(fmaxf(stats[8 + f0 + 0] * inv_n - mu * mu, 0.f)) + EPS_NORM;
        r.x = (v.x - mu) / sd;
    }
    {
        float mu = stats[f0 + 1] * inv_n;
        float sd = sqrtf(fmaxf(stats[8 + f0 + 1] * inv_n - mu * mu, 0.f)) + EPS_NORM;
        r.y = (v.y - mu) / sd;
    }
    {
        float mu = stats[f0 + 2] * inv_n;
        float sd = sqrtf(fmaxf(stats[8 + f0 + 2] * inv_n - mu * mu, 0.f)) + EPS_NORM;
        r.z = (v.z - mu) / sd;
    }
    {
        float mu = stats[f0 + 3] * inv_n;
        float sd = sqrtf(fmaxf(stats[8 + f0 + 3] * inv_n - mu * mu, 0.f)) + EPS_NORM;
        r.w = (v.w - mu) / sd;
    }
    *(fvec4*)(x + base) = r;
}

// ---- per-layer node-state init: m = -inf key, denom = 0, agg = 0 ----
__global__ __launch_bounds__(256) void k_node_init(int* __restrict__ mkey,
                                                   float* __restrict__ denom,
                                                   float* __restrict__ agg, int n) {
    int i = blockIdx.x * blockDim.x + threadIdx.x;
    if (i >= n) return;
    mkey[i]  = (int)0x80000000;   // key-space -infinity
    denom[i] = 0.f;
    const fvec4 z = {0.f, 0.f, 0.f, 0.f};
    *(fvec4*)(agg + i * N_FEAT)     = z;
    *(fvec4*)(agg + i * N_FEAT + 4) = z;
}

// ---- h = x @ W + b via V_WMMA_F32_16X16X4_F32 (one wave = 16 rows, K=8 as 2x K=4) ----
__global__ __launch_bounds__(256) void k_linear_wmma(const float* __restrict__ x,
                                                     const float* __restrict__ W,
                                                     const float* __restrict__ b,
                                                     float* __restrict__ h, int n) {
#if HAVE_WMMA_F32X4
    const int lane = threadIdx.x & 31;
    const int wave = blockIdx.x * (blockDim.x >> 5) + (threadIdx.x >> 5);
    const int row0 = wave * 16;
    if (row0 >= n) return;                    // wave-uniform exit; EXEC all-1s at WMMA
    const int m  = lane & 15;                 // A: row-in-tile; B/C/D: column
    const int kh = (lane >> 4) << 1;          // 0 (lanes 0-15) or 2 (lanes 16-31)
    const int row = row0 + m;                 // N_NODES % 16 == 0
    const float* xr = x + row * N_FEAT;
    // A 16x4 layout: lanes 0-15 hold K={0,1}, lanes 16-31 hold K={2,3}
    v2f a0, a1, b0, b1;
    a0.x = xr[kh + 0];     a0.y = xr[kh + 1];
    a1.x = xr[4 + kh + 0]; a1.y = xr[4 + kh + 1];
    // B 4x16: column = m (pad cols 8..15 with zeros), rows mirror A's K split
    const float sel = (m < N_FEAT) ? 1.f : 0.f;
    const int   col = (m < N_FEAT) ? m : 0;
    b0.x = W[(kh + 0) * N_FEAT + col] * sel;
    b0.y = W[(kh + 1) * N_FEAT + col] * sel;
    b1.x = W[(4 + kh + 0) * N_FEAT + col] * sel;
    b1.y = W[(4 + kh + 1) * N_FEAT + col] * sel;
    v8f c = {};
    c = __builtin_amdgcn_wmma_f32_16x16x4_f32(false, a0, false, b0, (short)0, c, false, false);
    c = __builtin_amdgcn_wmma_f32_16x16x4_f32(false, a1, false, b1, (short)0, c, false, false);
    // C/D: VGPR v -> row (v | v+8), lane -> column
    if (m < N_FEAT) {
        const float bias  = b[m];
        const int   rbase = row0 + ((lane < 16) ? 0 : 8);
        #pragma unroll
        for (int v = 0; v < 8; ++v)
            h[(rbase + v) * N_FEAT + m] = c[v] + bias;
    }
#else
    // scalar fallback (host pass / builtin missing): one thread per (row, feat)
    int i = blockIdx.x * blockDim.x + threadIdx.x;
    if (i >= n * N_FEAT) return;
    int row = i >> 3, f = i & 7;
    float acc = b[f];
    #pragma unroll
    for (int k = 0; k < N_FEAT; ++k) acc += x[row * N_FEAT + k] * W[k * N_FEAT + f];
    h[i] = acc;
#endif
}

// ---- edge logits + segment max over dst; 4 edges/thread ----
__global__ __launch_bounds__(256) void k_edge_logits(const int* __restrict__ esrc,
                                                     const int* __restrict__ edst,
                                                     const float* __restrict__ h,
                                                     const float* __restrict__ a_src,
                                                     const float* __restrict__ a_dst,
                                                     float* __restrict__ elog,
                                                     int* __restrict__ mkey, int ne4) {
    int t = blockIdx.x * blockDim.x + threadIdx.x;
    if (t >= ne4) return;
    const ivec4 s4 = __builtin_nontemporal_load((const ivec4*)(esrc) + t);
    const ivec4 d4 = __builtin_nontemporal_load((const ivec4*)(edst) + t);
    const fvec4 as0 = *(const fvec4*)(a_src);
    const fvec4 as1 = *(const fvec4*)(a_src + 4);
    const fvec4 ad0 = *(const fvec4*)(a_dst);
    const fvec4 ad1 = *(const fvec4*)(a_dst + 4);
    fvec4 lg;
    {
        const fvec4 hs0 = *(const fvec4*)(h + s4.x * N_FEAT);
        const fvec4 hs1 = *(const fvec4*)(h + s4.x * N_FEAT + 4);
        const fvec4 hd0 = *(const fvec4*)(h + d4.x * N_FEAT);
        const fvec4 hd1 = *(const fvec4*)(h + d4.x * N_FEAT + 4);
        float a = dot8(hs0, hs1, as0, as1) + dot8(hd0, hd1, ad0, ad1);
        lg.x = (a > 0.f) ? a : LEAKY * a;
        atomicMax(&mkey[d4.x], f2key(lg.x));
    }
    {
        const fvec4 hs0 = *(const fvec4*)(h + s4.y * N_FEAT);
        const fvec4 hs1 = *(const fvec4*)(h + s4.y * N_FEAT + 4);
        const fvec4 hd0 = *(const fvec4*)(h + d4.y * N_FEAT);
        const fvec4 hd1 = *(const fvec4*)(h + d4.y * N_FEAT + 4);
        float a = dot8(hs0, hs1, as0, as1) + dot8(hd0, hd1, ad0, ad1);
        lg.y = (a > 0.f) ? a : LEAKY * a;
        atomicMax(&mkey[d4.y], f2key(lg.y));
    }
    {
        const fvec4 hs0 = *(const fvec4*)(h + s4.z * N_FEAT);
        const fvec4 hs1 = *(const fvec4*)(h + s4.z * N_FEAT + 4);
        const fvec4 hd0 = *(const fvec4*)(h + d4.z * N_FEAT);
        const fvec4 hd1 = *(const fvec4*)(h + d4.z * N_FEAT + 4);
        float a = dot8(hs0, hs1, as0, as1) + dot8(hd0, hd1, ad0, ad1);
        lg.z = (a > 0.f) ? a : LEAKY * a;
        atomicMax(&mkey[d4.z], f2key(lg.z));
    }
    {
        const fvec4 hs0 = *(const fvec4*)(h + s4.w * N_FEAT);
        const fvec4 hs1 = *(const fvec4*)(h + s4.w * N_FEAT + 4);
        const fvec4 hd0 = *(const fvec4*)(h + d4.w * N_FEAT);
        const fvec4 hd1 = *(const fvec4*)(h + d4.w * N_FEAT + 4);
        float a = dot8(hs0, hs1, as0, as1) + dot8(hd0, hd1, ad0, ad1);
        lg.w = (a > 0.f) ? a : LEAKY * a;
        atomicMax(&mkey[d4.w], f2key(lg.w));
    }
    *((fvec4*)elog + t) = lg;
}

// ---- fused: e = exp(logit - m[dst]); denom[dst] += e; agg[dst] += e*h[src] ----
// (agg is un-normalized; k_norm_elu divides by denom afterwards — algebraically
//  identical to the reference's per-edge w = e/denom weighting)
__global__ __launch_bounds__(256) void k_edge_expagg(const int* __restrict__ esrc,
                                                     const int* __restrict__ edst,
                                                     const float* __restrict__ h,
                                                     const float* __restrict__ elog,
                                                     const int* __restrict__ mkey,
                                                     float* __restrict__ denom,
                                                     float* __restrict__ agg, int ne4) {
    int t = blockIdx.x * blockDim.x + threadIdx.x;
    if (t >= ne4) return;
    const ivec4 s4 = __builtin_nontemporal_load((const ivec4*)(esrc) + t);
    const ivec4 d4 = __builtin_nontemporal_load((const ivec4*)(edst) + t);
    const fvec4 lg = *((const fvec4*)elog + t);
    {
        const float e = expf(lg.x - key2f(mkey[d4.x]));
        atomicAdd(&denom[d4.x], e);
        const fvec4 h0 = *(const fvec4*)(h + s4.x * N_FEAT);
        const fvec4 h1 = *(const fvec4*)(h + s4.x * N_FEAT + 4);
        float* a = agg + d4.x * N_FEAT;
        atomicAdd(a + 0, e * h0.x); atomicAdd(a + 1, e * h0.y);
        atomicAdd(a + 2, e * h0.z); atomicAdd(a + 3, e * h0.w);
        atomicAdd(a + 4, e * h1.x); atomicAdd(a + 5, e * h1.y);
        atomicAdd(a + 6, e * h1.z); atomicAdd(a + 7, e * h1.w);
    }
    {
        const float e = expf(lg.y - key2f(mkey[d4.y]));
        atomicAdd(&denom[d4.y], e);
        const fvec4 h0 = *(const fvec4*)(h + s4.y * N_FEAT);
        const fvec4 h1 = *(const fvec4*)(h + s4.y * N_FEAT + 4);
        float* a = agg + d4.y * N_FEAT;
        atomicAdd(a + 0, e * h0.x); atomicAdd(a + 1, e * h0.y);
        atomicAdd(a + 2, e * h0.z); atomicAdd(a + 3, e * h0.w);
        atomicAdd(a + 4, e * h1.x); atomicAdd(a + 5, e * h1.y);
        atomicAdd(a + 6, e * h1.z); atomicAdd(a + 7, e * h1.w);
    }
    {
        const float e = expf(lg.z - key2f(mkey[d4.z]));
        atomicAdd(&denom[d4.z], e);
        const fvec4 h0 = *(const fvec4*)(h + s4.z * N_FEAT);
        const fvec4 h1 = *(const fvec4*)(h + s4.z * N_FEAT + 4);
        float* a = agg + d4.z * N_FEAT;
        atomicAdd(a + 0, e * h0.x); atomicAdd(a + 1, e * h0.y);
        atomicAdd(a + 2, e * h0.z); atomicAdd(a + 3, e * h0.w);
        atomicAdd(a + 4, e * h1.x); atomicAdd(a + 5, e * h1.y);
        atomicAdd(a + 6, e * h1.z); atomicAdd(a + 7, e * h1.w);
    }
    {
        const float e = expf(lg.w - key2f(mkey[d4.w]));
        atomicAdd(&denom[d4.w], e);
        const fvec4 h0 = *(const fvec4*)(h + s4.w * N_FEAT);
        const fvec4 h1 = *(const fvec4*)(h + s4.w * N_FEAT + 4);
        float* a = agg + d4.w * N_FEAT;
        atomicAdd(a + 0, e * h0.x); atomicAdd(a + 1, e * h0.y);
        atomicAdd(a + 2, e * h0.z); atomicAdd(a + 3, e * h0.w);
        atomicAdd(a + 4, e * h1.x); atomicAdd(a + 5, e * h1.y);
        atomicAdd(a + 6, e * h1.z); atomicAdd(a + 7, e * h1.w);
    }
}

// ---- x = elu(agg / (denom + 1e-9)); 4 elems/thread (one node half-row) ----
__global__ __launch_bounds__(256) void k_norm_elu(const float* __restrict__ agg,
                                                  const float* __restrict__ denom,
                                                  float* __restrict__ x, int total4) {
    int t = blockIdx.x * blockDim.x + threadIdx.x;
    if (t >= total4) return;
    const int base = t * 4;
    const int node = base >> 3;               // 4-aligned chunk stays within one node row
    const float inv = 1.f / (denom[node] + 1e-9f);
    fvec4 a = *(const fvec4*)(agg + base);
    fvec4 r;
    a.x *= inv; a.y *= inv; a.z *= inv; a.w *= inv;
    r.x = (a.x > 0.f) ? a.x : expm1f(a.x);
    r.y = (a.y > 0.f) ? a.y : expm1f(a.y);
    r.z = (a.z > 0.f) ? a.z : expm1f(a.z);
    r.w = (a.w > 0.f) ? a.w : expm1f(a.w);
    *(fvec4*)(x + base) = r;
}

// ---- edge head: mask + linear on concat(x[src], x[dst]); 4 edges/thread ----
__global__ __launch_bounds__(256) void k_edge_val(const int* __restrict__ esrc,
                                                  const int* __restrict__ edst,
                                                  const float* __restrict__ xin,
                                                  const float* __restrict__ x,
                                                  const float* __restrict__ evW,
                                                  const float* __restrict__ evb,
                                                  float* __restrict__ out_edge,
                                                  float* __restrict__ out_mask, int ne4) {
    int t = blockIdx.x * blockDim.x + threadIdx.x;
    if (t >= ne4) return;
    const ivec4 s4 = __builtin_nontemporal_load((const ivec4*)(esrc) + t);
    const ivec4 d4 = __builtin_nontemporal_load((const ivec4*)(edst) + t);
    const fvec4 ws0 = *(const fvec4*)(evW);
    const fvec4 ws1 = *(const fvec4*)(evW + 4);
    const fvec4 wd0 = *(const fvec4*)(evW + 8);
    const fvec4 wd1 = *(const fvec4*)(evW + 12);
    const float bias = evb[0];
    fvec4 val, msk;
    {
        const int s = s4.x, d = d4.x;
        bool act = (xin[s * N_FEAT + 0] > 0.f) && (xin[s * N_FEAT + 1] > 1.f) &&
                   (xin[d * N_FEAT + 0] <= 0.f);
        const fvec4 xs0 = *(const fvec4*)(x + s * N_FEAT);
        const fvec4 xs1 = *(const fvec4*)(x + s * N_FEAT + 4);
        const fvec4 xd0 = *(const fvec4*)(x + d * N_FEAT);
        const fvec4 xd1 = *(const fvec4*)(x + d * N_FEAT + 4);
        float a = bias + dot8(xs0, xs1, ws0, ws1) + dot8(xd0, xd1, wd0, wd1);
        val.x = act ? a : 0.f;  msk.x = act ? 1.f : 0.f;
    }
    {
        const int s = s4.y, d = d4.y;
        bool act = (xin[s * N_FEAT + 0] > 0.f) && (xin[s * N_FEAT + 1] > 1.f) &&
                   (xin[d * N_FEAT + 0] <= 0.f);
        const fvec4 xs0 = *(const fvec4*)(x + s * N_FEAT);
        const fvec4 xs1 = *(const fvec4*)(x + s * N_FEAT + 4);
        const fvec4 xd0 = *(const fvec4*)(x + d * N_FEAT);
        const fvec4 xd1 = *(const fvec4*)(x + d * N_FEAT + 4);
        float a = bias + dot8(xs0, xs1, ws0, ws1) + dot8(xd0, xd1, wd0, wd1);
        val.y = act ? a : 0.f;  msk.y = act ? 1.f : 0.f;
    }
    {
        const int s = s4.z, d = d4.z;
        bool act = (xin[s * N_FEAT + 0] > 0.f) && (xin[s * N_FEAT + 1] > 1.f) &&
                   (xin[d * N_FEAT + 0] <= 0.f);
        const fvec4 xs0 = *(const fvec4*)(x + s * N_FEAT);
        const fvec4 xs1 = *(const fvec4*)(x + s * N_FEAT + 4);
        const fvec4 xd0 = *(const fvec4*)(x + d * N_FEAT);
        const fvec4 xd1 = *(const fvec4*)(x + d * N_FEAT + 4);
        float a = bias + dot8(xs0, xs1, ws0, ws1) + dot8(xd0, xd1, wd0, wd1);
        val.z = act ? a : 0.f;  msk.z = act ? 1.f : 0.f;
    }
    {
        const int s = s4.w, d = d4.w;
        bool act = (xin[s * N_FEAT + 0] > 0.f) && (xin[s * N_FEAT + 1] > 1.f) &&
                   (xin[d * N_FEAT + 0] <= 0.f);
        const fvec4 xs0 = *(const fvec4*)(x + s * N_FEAT);
        const fvec4 xs1 = *(const fvec4*)(x + s * N_FEAT + 4);
        const fvec4 xd0 = *(const fvec4*)(x + d * N_FEAT);
        const fvec4 xd1 = *(const fvec4*)(x + d * N_FEAT + 4);
        float a = bias + dot8(xs0, xs1, ws0, ws1) + dot8(xd0, xd1, wd0, wd1);
        val.w = act ? a : 0.f;  msk.w = act ? 1.f : 0.f;
    }
    __builtin_nontemporal_store(val, (fvec4*)(out_edge) + t);
    __builtin_nontemporal_store(msk, (fvec4*)(out_mask) + t);
}

// ---- per-graph pool: LDS-local sums (ids sorted -> low global contention) ----
__global__ __launch_bounds__(256) void k_pool(const float* __restrict__ x,
                                              const int* __restrict__ gid,
                                              float* __restrict__ gsum,
                                              float* __restrict__ gcnt, int n) {
    __shared__ float ls[NUM_GRAPHS * N_FEAT];
    __shared__ float lc[NUM_GRAPHS];
    for (int i = threadIdx.x; i < NUM_GRAPHS * N_FEAT; i += blockDim.x) ls[i] = 0.f;
    for (int i = threadIdx.x; i < NUM_GRAPHS; i += blockDim.x) lc[i] = 0.f;
    __syncthreads();
    int i = blockIdx.x * blockDim.x + threadIdx.x;
    if (i < n) {
        int g = gid[i];
        const fvec4 v0 = *(const fvec4*)(x + i * N_FEAT);
        const fvec4 v1 = *(const fvec4*)(x + i * N_FEAT + 4);
        float* l = ls + g * N_FEAT;
        atomicAdd(l + 0, v0.x); atomicAdd(l + 1, v0.y);
        atomicAdd(l + 2, v0.z); atomicAdd(l + 3, v0.w);
        atomicAdd(l + 4, v1.x); atomicAdd(l + 5, v1.y);
        atomicAdd(l + 6, v1.z); atomicAdd(l + 7, v1.w);
        atomicAdd(&lc[g], 1.f);
    }
    __syncthreads();
    for (int i2 = threadIdx.x; i2 < NUM_GRAPHS * N_FEAT; i2 += blockDim.x)
        if (ls[i2] != 0.f) atomicAdd(&gsum[i2], ls[i2]);
    for (int i2 = threadIdx.x; i2 < NUM_GRAPHS; i2 += blockDim.x)
        if (lc[i2] != 0.f) atomicAdd(&gcnt[i2], lc[i2]);
}

// ---- graph head: out[g] = mean(x over graph) @ gvW + gvb ----
__global__ void k_head(const float* __restrict__ gsum, const float* __restrict__ gcnt,
                       const float* __restrict__ W, const float* __restrict__ b,
                       float* __restrict__ out) {
    int g = threadIdx.x;
    if (g >= NUM_GRAPHS) return;
    float inv = 1.f / fmaxf(gcnt[g], 1.f);
    float acc = b[0];
    #pragma unroll
    for (int f = 0; f < N_FEAT; ++f) acc += gsum[g * N_FEAT + f] * inv * W[f];
    out[g] = acc;
}

extern "C" void kernel_launch(void* const* d_in, const int* in_sizes, int n_in,
                              void* d_out, int out_size, void* d_ws, size_t ws_size,
                              hipStream_t stream) {
    // Input order: setup_inputs() dict order, then params as tree leaves
    // (dict keys sorted: edge_val{W,b}, gat[0..2]{a_dst,a_src,lin{W,b}}, graph_val{W,b})
    const float* in_states = (const float*)d_in[0];
    const int*   edges     = (const int*)d_in[1];       // [2, E] flat: src then dst
    const int*   node_gid  = (const int*)d_in[2];
    /* d_in[3] edges_graph_ids: unused by reference output */
    const float* evW = (const float*)d_in[4];           // [16,1]
    const float* evb = (const float*)d_in[5];           // [1]
    const float* a_dst_l[3], *a_src_l[3], *W_l[3], *b_l[3];
    for (int l = 0; l < 3; ++l) {
        a_dst_l[l] = (const float*)d_in[6 + 4 * l + 0];
        a_src_l[l] = (const float*)d_in[6 + 4 * l + 1];
        W_l[l]     = (const float*)d_in[6 + 4 * l + 2]; // [8,8] row-major
        b_l[l]     = (const float*)d_in[6 + 4 * l + 3];
    }
    const float* gvW = (const float*)d_in[18];          // [8,1]
    const float* gvb = (const float*)d_in[19];          // [1]

    const int* esrc = edges;
    const int* edst = edges + N_EDGES;

    float* out_end  = (float*)d_out;                    // [64]
    float* out_edge = out_end + NUM_GRAPHS;             // [E]  (256B offset -> f4 aligned)
    float* out_mask = out_edge + N_EDGES;               // [E] bool as 0/1 float (16B aligned)

    // workspace carve-out (256B aligned)
    char* p = (char*)d_ws;
    auto carve = [&](size_t bytes) { void* r = (void*)p; p += (bytes + 255) & ~(size_t)255; return r; };
    float* x     = (float*)carve((size_t)N_NODES * N_FEAT * sizeof(float));
    float* h     = (float*)carve((size_t)N_NODES * N_FEAT * sizeof(float));
    float* agg   = (float*)carve((size_t)N_NODES * N_FEAT * sizeof(float));
    int*   mkey  = (int*)  carve((size_t)N_NODES * sizeof(int));
    float* denom = (float*)carve((size_t)N_NODES * sizeof(float));
    float* elog  = (float*)carve((size_t)N_EDGES * sizeof(float));
    float* stats = (float*)carve(16 * sizeof(float));
    float* gsum  = (float*)carve(NUM_GRAPHS * N_FEAT * sizeof(float));
    float* gcnt  = (float*)carve(NUM_GRAPHS * sizeof(float));

    const int TB   = 256;
    const int NB   = (N_NODES + TB - 1) / TB;                    // node-grain blocks
    const int NF4  = N_NODES * N_FEAT / 4;                       // 4-elem chunks
    const int NF4B = (NF4 + TB - 1) / TB;
    const int NFB  = (N_NODES * N_FEAT + TB - 1) / TB;           // scalar-fallback cover
    const int E4   = N_EDGES / 4;                                // 4 edges per thread
    const int E4B  = (E4 + TB - 1) / TB;

    k_init_small<<<1, 512, 0, stream>>>(stats, gsum, gcnt);
    k_stats<<<NB, TB, 0, stream>>>(in_states, stats, N_NODES);
    k_dwnorm<<<NF4B, TB, 0, stream>>>(in_states, stats, x, NF4);

    for (int l = 0; l < 3; ++l) {
        k_node_init<<<NB, TB, 0, stream>>>(mkey, denom, agg, N_NODES);
        // NFB blocks covers both the WMMA path (needs ceil(N/16/8)=782) and fallback
        k_linear_wmma<<<NFB, TB, 0, stream>>>(x, W_l[l], b_l[l], h, N_NODES);
        k_edge_logits<<<E4B, TB, 0, stream>>>(esrc, edst, h, a_src_l[l], a_dst_l[l],
                                              elog, mkey, E4);
        k_edge_expagg<<<E4B, TB, 0, stream>>>(esrc, edst, h, elog, mkey,
                                              denom, agg, E4);
        k_norm_elu<<<NF4B, TB, 0, stream>>>(agg, denom, x, NF4);
    }

    k_edge_val<<<E4B, TB, 0, stream>>>(esrc, edst, in_states, x, evW, evb,
                                       out_edge, out_mask, E4);
    k_pool<<<NB, TB, 0, stream>>>(x, node_gid, gsum, gcnt, N_NODES);
    k_head<<<1, 64, 0, stream>>>(gsum, gcnt, gvW, gvb, out_end);
}